// IOU_SSIM_46531675685416
// MI455X (gfx1250) — compile-verified
//
#include <hip/hip_runtime.h>

// ---------------- problem constants ----------------
#define SZ   32
#define NB   8      // batch
#define NG   16     // gt boxes per batch
#define NP   64     // pred boxes per batch
#define IMH  640
#define IMW  640
#define C1K  6.5025f    // (0.01*255)^2
#define C2K  58.5225f   // (0.03*255)^2

// normalized 1D Gaussian, sigma=1.5, width 11
__device__ __constant__ float GW[11] = {
    0.00102838f, 0.00759875f, 0.03600077f, 0.10936069f, 0.21300553f,
    0.26601172f, 0.21300553f, 0.10936069f, 0.03600077f, 0.00759875f,
    0.00102838f};

// ---------------- workspace layout (floats) ----------------
constexpr size_t NGT   = (size_t)NB * NG * 3 * 1024;   // gt channel-patches
constexpr size_t NPR   = (size_t)NB * NP * 3 * 1024;   // pred channel-patches
constexpr size_t OFF_GT   = 0;
constexpr size_t OFF_PR   = OFF_GT  + NGT;
constexpr size_t OFF_MUG  = OFF_PR  + NPR;
constexpr size_t OFF_E2G  = OFF_MUG + NGT;
constexpr size_t OFF_MUP  = OFF_E2G + NGT;
constexpr size_t OFF_E2P  = OFF_MUP + NPR;
constexpr size_t OFF_MASK = OFF_E2P + NPR;
constexpr size_t OFF_ACC  = OFF_MASK + (size_t)NB * NG * NP;  // [total, cnt]

typedef float v2f __attribute__((ext_vector_type(2)));
typedef float v8f __attribute__((ext_vector_type(8)));

// ---------------- wave-level 32x32 f32 GEMM via V_WMMA_F32_16X16X4_F32 ----
// C(32x32) = A(32x32) * B(32x32).  One wave32.  A/B/C generic pointers,
// row stride 32.  Optionally squares A elements on read (for E[x^2] conv).
__device__ __forceinline__ void wave_gemm32(const float* A, const float* B,
                                            float* C, int lane, bool squareA) {
  const int n    = lane & 15;   // column / M-row selector within tile
  const int half = lane >> 4;   // K-pair selector (0 -> K{0,1}, 1 -> K{2,3})
  #pragma unroll
  for (int tm = 0; tm < 32; tm += 16) {
    #pragma unroll
    for (int tn = 0; tn < 32; tn += 16) {
      v8f acc = {0.f, 0.f, 0.f, 0.f, 0.f, 0.f, 0.f, 0.f};
      #pragma unroll
      for (int kc = 0; kc < 32; kc += 4) {
        const int k0 = kc + 2 * half;
        v2f a, b;
        a.x = A[(tm + n) * 32 + k0];
        a.y = A[(tm + n) * 32 + k0 + 1];
        if (squareA) { a.x *= a.x; a.y *= a.y; }
        b.x = B[(k0)     * 32 + tn + n];
        b.y = B[(k0 + 1) * 32 + tn + n];
        acc = __builtin_amdgcn_wmma_f32_16x16x4_f32(
            false, a, false, b, (short)0, acc, false, false);
      }
      #pragma unroll
      for (int r = 0; r < 8; ++r)
        C[(tm + r + 8 * half) * 32 + tn + n] = acc[r];
    }
  }
}

// band matrix G[k][j] = GW[j-k+5] for |j-k|<=5 (symmetric -> conv = G*X*G)
__device__ __forceinline__ void build_G(float* Gm, int lane) {
  const int k = lane;  // one row per lane (wave32)
  #pragma unroll
  for (int j = 0; j < 32; ++j) {
    int d = j - k;
    Gm[k * 32 + j] = (d >= -5 && d <= 5) ? GW[d + 5] : 0.f;
  }
}

// ---------------- kernel 0: zero accumulators ----------------
__global__ void init_kernel(float* acc) { acc[threadIdx.x] = 0.f; }

// ---------------- kernel 1: bilinear crop-resize ----------------
__global__ void crop_kernel(const float* __restrict__ imgs,
                            const float* __restrict__ boxes,
                            float* __restrict__ out, int nbox) {
  const int blk = blockIdx.x;
  const int b = blk / nbox, k = blk % nbox;
  const float* box = boxes + (size_t)(b * nbox + k) * 4;
  const float bx = box[0], by = box[1], bw = box[2], bh = box[3];
  const float x0 = floorf(bx), y0 = floorf(by);
  const float w  = floorf(bx + bw) - x0;
  const float h  = floorf(by + bh) - y0;
  const float hm1 = fmaxf(h - 1.f, 0.f), wm1 = fmaxf(w - 1.f, 0.f);
  const float* img = imgs + (size_t)b * 3 * IMH * IMW;
  float* dst = out + (size_t)(b * nbox + k) * 3 * SZ * SZ;
  for (int e = threadIdx.x; e < 3 * SZ * SZ; e += blockDim.x) {
    const int c = e >> 10, rem = e & 1023, i = rem >> 5, j = rem & 31;
    float ys = fminf(fmaxf((i + 0.5f) * (h * (1.f / SZ)) - 0.5f, 0.f), hm1);
    float xs = fminf(fmaxf((j + 0.5f) * (w * (1.f / SZ)) - 0.5f, 0.f), wm1);
    float yf = floorf(ys), xf = floorf(xs);
    float wy = ys - yf, wx = xs - xf;
    float ylo = yf + y0, yhi = fminf(yf + 1.f, hm1) + y0;
    float xlo = xf + x0, xhi = fminf(xf + 1.f, wm1) + x0;
    const float* ic = img + (size_t)c * IMH * IMW;
    auto gi = [&](float yy, float xx) -> float {
      int yi = (int)fminf(fmaxf(yy, 0.f), (float)(IMH - 1));
      int xi = (int)fminf(fmaxf(xx, 0.f), (float)(IMW - 1));
      return ic[(size_t)yi * IMW + xi];
    };
    float top = gi(ylo, xlo) * (1.f - wx) + gi(ylo, xhi) * wx;
    float bot = gi(yhi, xlo) * (1.f - wx) + gi(yhi, xhi) * wx;
    dst[e] = top * (1.f - wy) + bot * wy;
  }
}

// ---------------- kernel 2: pairwise IoU -> mask + count ----------------
__global__ void iou_kernel(const float* __restrict__ gt,
                           const float* __restrict__ pr,
                           float* __restrict__ mask, float* __restrict__ cnt) {
  const int idx = blockIdx.x * blockDim.x + threadIdx.x;
  if (idx >= NB * NG * NP) return;
  const int b = idx / (NG * NP), rem = idx % (NG * NP);
  const int g = rem / NP, p = rem % NP;
  const float* gb = gt + (size_t)(b * NG + g) * 4;
  const float* pb = pr + (size_t)(b * NP + p) * 4;
  const float gx = gb[0], gy = gb[1], gw = gb[2], gh = gb[3];
  const float px = pb[0], py = pb[1], pw = pb[2], ph = pb[3];
  float tlx = fmaxf(gx - gw * 0.5f, px - pw * 0.5f);
  float tly = fmaxf(gy - gh * 0.5f, py - ph * 0.5f);
  float brx = fminf(gx + gw * 0.5f, px + pw * 0.5f);
  float bry = fminf(gy + gh * 0.5f, py + ph * 0.5f);
  float en = (tlx < brx && tly < bry) ? 1.f : 0.f;
  float ai = (brx - tlx) * (bry - tly) * en;
  float iou = ai / (gw * gh + pw * ph - ai + 1e-16f);
  float m = (iou > 0.3f && pw > 2.f && ph > 2.f) ? 1.f : 0.f;
  mask[idx] = m;
  if (m > 0.f) atomicAdd(cnt, 1.f);
}

// ---------------- kernel 3: per-patch mu = G*X*G, e2 = G*(X.X)*G ----------
__global__ void patchstat_kernel(float* __restrict__ ws) {
  __shared__ float Gm[1024], S[1024], T[1024];
  const int lane = threadIdx.x;
  const int pi = blockIdx.x;  // channel-patch index over gt then pred
  const float* src; float* mu; float* e2;
  if (pi < NB * NG * 3) {
    src = ws + OFF_GT  + (size_t)pi * 1024;
    mu  = ws + OFF_MUG + (size_t)pi * 1024;
    e2  = ws + OFF_E2G + (size_t)pi * 1024;
  } else {
    const int q = pi - NB * NG * 3;
    src = ws + OFF_PR  + (size_t)q * 1024;
    mu  = ws + OFF_MUP + (size_t)q * 1024;
    e2  = ws + OFF_E2P + (size_t)q * 1024;
  }
  build_G(Gm, lane);
  for (int e = lane; e < 1024; e += 32) S[e] = src[e];
  __syncthreads();
  wave_gemm32(S, Gm, T, lane, false);   // T = X*G
  __syncthreads();
  wave_gemm32(Gm, T, mu, lane, false);  // mu = G*T  (to global)
  __syncthreads();
  wave_gemm32(S, Gm, T, lane, true);    // T = (X.X)*G
  __syncthreads();
  wave_gemm32(Gm, T, e2, lane, false);  // e2 = G*T  (to global)
}

// ---------------- kernel 4: per-pair SSIM + L1, masked ----------------
__global__ void pair_kernel(float* __restrict__ ws, float* __restrict__ total) {
  __shared__ float Gm[1024], X[1024], Y[1024], XY[1024], T[1024], O[1024];
  const int lane = threadIdx.x;
  const int idx = blockIdx.x;
  if (ws[OFF_MASK + idx] == 0.f) return;   // wave-uniform early out
  const int b = idx / (NG * NP), rem = idx % (NG * NP);
  const int g = rem / NP, p = rem % NP;
  build_G(Gm, lane);
  float lsum = 0.f;
  for (int c = 0; c < 3; ++c) {
    const size_t prb = (size_t)((b * NP + p) * 3 + c) * 1024;
    const size_t gtb = (size_t)((b * NG + g) * 3 + c) * 1024;
    const float* xs = ws + OFF_PR + prb;
    const float* yr = ws + OFF_GT + gtb;
    for (int e = lane; e < 1024; e += 32) {
      float xv = xs[e], yv = yr[e];
      X[e] = xv; Y[e] = yv; XY[e] = xv * yv;
    }
    __syncthreads();
    wave_gemm32(XY, Gm, T, lane, false);  // T = (X.Y)*G
    __syncthreads();
    wave_gemm32(Gm, T, O, lane, false);   // O = gconv(x*y)
    __syncthreads();
    const float* mu1p = ws + OFF_MUP + prb;
    const float* e2xp = ws + OFF_E2P + prb;
    const float* mu2p = ws + OFF_MUG + gtb;
    const float* e2yp = ws + OFF_E2G + gtb;
    for (int e = lane; e < 1024; e += 32) {
      float mu1 = mu1p[e], mu2 = mu2p[e];
      float mu1s = mu1 * mu1, mu2s = mu2 * mu2, mu12 = mu1 * mu2;
      float s1 = e2xp[e] - mu1s;
      float s2 = e2yp[e] - mu2s;
      float s12 = O[e] - mu12;
      float ssim = ((2.f * mu12 + C1K) * (2.f * s12 + C2K)) /
                   ((mu1s + mu2s + C1K) * (s1 + s2 + C2K));
      lsum += fabsf(X[e] - Y[e]) * (1.f / 255.f) + (1.f - ssim);
    }
    __syncthreads();
  }
  #pragma unroll
  for (int off = 16; off > 0; off >>= 1) lsum += __shfl_xor(lsum, off, 32);
  if (lane == 0) atomicAdd(total, lsum);
}

// ---------------- kernel 5: finalize scalar ----------------
__global__ void finalize_kernel(const float* __restrict__ acc,
                                float* __restrict__ out) {
  const float total = acc[0];
  const float cnt = acc[1] * (float)(3 * SZ * SZ);
  out[0] = (cnt > 0.f) ? total / fmaxf(cnt, 1.f) : 0.f;
}

// ---------------- host launcher ----------------
extern "C" void kernel_launch(void* const* d_in, const int* in_sizes, int n_in,
                              void* d_out, int out_size, void* d_ws,
                              size_t ws_size, hipStream_t stream) {
  const float* gt   = (const float*)d_in[0];  // (8,16,4)
  const float* pr   = (const float*)d_in[1];  // (8,64,4)
  const float* imgs = (const float*)d_in[2];  // (8,3,640,640)
  float* ws  = (float*)d_ws;
  float* out = (float*)d_out;
  float* acc = ws + OFF_ACC;  // [total, cnt]

  init_kernel<<<1, 2, 0, stream>>>(acc);
  crop_kernel<<<NB * NG, 128, 0, stream>>>(imgs, gt, ws + OFF_GT, NG);
  crop_kernel<<<NB * NP, 128, 0, stream>>>(imgs, pr, ws + OFF_PR, NP);
  iou_kernel<<<(NB * NG * NP + 255) / 256, 256, 0, stream>>>(
      gt, pr, ws + OFF_MASK, acc + 1);
  patchstat_kernel<<<(NB * NG + NB * NP) * 3, 32, 0, stream>>>(ws);
  pair_kernel<<<NB * NG * NP, 32, 0, stream>>>(ws, acc);
  finalize_kernel<<<1, 1, 0, stream>>>(acc, out);
}